// BiologyEncoder_34900904247632
// MI455X (gfx1250) — compile-verified
//
#include <hip/hip_runtime.h>

// ---------------------------------------------------------------------------
// GCN encoder for MI455X (gfx1250, wave32).
// Dense GEMMs use V_WMMA_F32_16X16X4_F32 (full f32, matches reference dtype;
// GEMMs are compute-trivial vs the L2-bound edge scatter, so no need for bf16).
// Edge aggregation: self-loop folded analytically, edges via coalesced float4
// gathers + global_atomic_add_f32 (L2-resident: working set << 192MB L2).
// Segment-max uses integer atomicMax on float bits (valid: post-relu h >= 0).
// ---------------------------------------------------------------------------

typedef __attribute__((ext_vector_type(2))) float v2f;
typedef __attribute__((ext_vector_type(8))) float v8f;

#define N_NODES  50000
#define N_EDGES  800000
#define N_GRAPHS 64

__global__ void fill_f32(float* __restrict__ p, float v, int n) {
    int i = blockIdx.x * blockDim.x + threadIdx.x;
    if (i < n) p[i] = v;
}

__global__ void edge_deg(const long long* __restrict__ dst, float* __restrict__ deg, int e) {
    int i = blockIdx.x * blockDim.x + threadIdx.x;
    if (i < e) atomicAdd(&deg[(int)dst[i]], 1.0f);
}

__global__ void to_rsqrt(float* __restrict__ p, int n) {
    int i = blockIdx.x * blockDim.x + threadIdx.x;
    if (i < n) p[i] = rsqrtf(p[i]);
}

// h0[n,64] = x[n,6] @ W_in[6,64] + b_in   (tiny: 38 MFLOP, VALU is fine)
__global__ void input_proj(const float* __restrict__ x, const float* __restrict__ W,
                           const float* __restrict__ b, float* __restrict__ h, int n) {
    __shared__ float Ws[6 * 64];
    for (int i = threadIdx.x; i < 6 * 64; i += blockDim.x) Ws[i] = W[i];
    __syncthreads();
    int idx = blockIdx.x * blockDim.x + threadIdx.x;
    if (idx >= n * 64) return;
    int node = idx >> 6, col = idx & 63;
    const float* xr = x + node * 6;
    float acc = b[col];
#pragma unroll
    for (int k = 0; k < 6; ++k) acc += xr[k] * Ws[k * 64 + col];
    h[idx] = acc;
}

// C[rows,NC] = A[rows,K] @ B[K,NC] (+bias)(+relu) via V_WMMA_F32_16X16X4_F32.
// One wave -> one 16x16 tile. Block = 128 threads = 4 waves = 4 column tiles.
// grid = (rows/16, NC/64). rows, NC, K are exact multiples (50000=16*3125).
// A frag (16x4): lane l -> row = l%16; VGPR v holds K = 2*(l/16) + v.
// B frag (4x16): lane l -> col = l%16; VGPR v holds K = 2*(l/16) + v.
// D (16x16):     lane l, VGPR r -> row = r + 8*(l/16), col = l%16.
__global__ void wmma_gemm_f32(const float* __restrict__ A, const float* __restrict__ B,
                              const float* __restrict__ bias, float* __restrict__ C,
                              int K, int NC, int relu) {
    int lane = threadIdx.x & 31;
    int wv   = threadIdx.x >> 5;
    int row0 = blockIdx.x * 16;
    int col0 = (blockIdx.y * 4 + wv) * 16;
    int m  = lane & 15;
    int hi = lane >> 4;                    // 0: K pair {0,1}, 1: K pair {2,3}
    const float* arow = A + (size_t)(row0 + m) * K;
    const float* bcol = B + col0 + m;
    v8f c = {0.f, 0.f, 0.f, 0.f, 0.f, 0.f, 0.f, 0.f};
    for (int kk = 0; kk < K; kk += 4) {
        int k0 = kk + 2 * hi;
        v2f a = { arow[k0], arow[k0 + 1] };
        v2f b = { bcol[(size_t)k0 * NC], bcol[(size_t)(k0 + 1) * NC] };
        c = __builtin_amdgcn_wmma_f32_16x16x4_f32(false, a, false, b, (short)0, c,
                                                  false, false);
    }
    float bv = bias ? bias[col0 + m] : 0.0f;
#pragma unroll
    for (int r = 0; r < 8; ++r) {
        float v = c[r] + bv;
        if (relu) v = fmaxf(v, 0.0f);
        C[(size_t)(row0 + hi * 8 + r) * NC + col0 + m] = v;
    }
}

// out[n,m] = b[m] + hw[n,m] * dinv[n]^2   (self-loop contribution, no atomics)
__global__ void init_agg(float* __restrict__ out, const float* __restrict__ hw,
                         const float* __restrict__ bias, const float* __restrict__ dinv,
                         int total) {
    int i = blockIdx.x * blockDim.x + threadIdx.x;
    if (i >= total) return;
    int n = i >> 7, mcol = i & 127;
    float di = dinv[n];
    out[i] = bias[mcol] + hw[i] * di * di;
}

// 32 consecutive threads per edge; each thread gathers float4 and does 4
// f32 atomic adds -> coalesced 512B row gather, L2-resident atomics.
__global__ void edge_scatter(const long long* __restrict__ src, const long long* __restrict__ dst,
                             const float* __restrict__ dinv, const float* __restrict__ hw,
                             float* __restrict__ out, int e) {
    int gid = blockIdx.x * blockDim.x + threadIdx.x;
    int ed = gid >> 5, sub = gid & 31;
    if (ed >= e) return;
    int s = (int)src[ed], d = (int)dst[ed];
    float nm = dinv[s] * dinv[d];
    const float4 v = *(const float4*)(hw + (size_t)s * 128 + sub * 4);
    float* dp = out + (size_t)d * 128 + sub * 4;
    atomicAdd(dp + 0, v.x * nm);
    atomicAdd(dp + 1, v.y * nm);
    atomicAdd(dp + 2, v.z * nm);
    atomicAdd(dp + 3, v.w * nm);
}

__global__ void relu_ip(float* __restrict__ p, int n) {
    int i = blockIdx.x * blockDim.x + threadIdx.x;
    if (i < n) p[i] = fmaxf(p[i], 0.0f);
}

// Segment sum/max pooling. h >= 0 post-relu, so uint atomicMax on float bits
// is order-preserving; gmax zero-init matches reference's where(counts>0,...,0).
__global__ void pool(const float* __restrict__ h, const long long* __restrict__ batch,
                     float* __restrict__ gsum, float* __restrict__ gmax,
                     float* __restrict__ gcnt, int total) {
    int i = blockIdx.x * blockDim.x + threadIdx.x;
    if (i >= total) return;
    int n = i >> 7, mcol = i & 127;
    int g = (int)batch[n];
    float v = h[i];
    atomicAdd(&gsum[g * 128 + mcol], v);
    atomicMax((unsigned int*)&gmax[g * 128 + mcol], __float_as_uint(v));
    if (mcol == 0) atomicAdd(&gcnt[g], 1.0f);
}

__global__ void finalize_hg(const float* __restrict__ gsum, const float* __restrict__ gmax,
                            const float* __restrict__ gcnt, float* __restrict__ hg) {
    int i = blockIdx.x * blockDim.x + threadIdx.x;
    if (i >= N_GRAPHS * 128) return;
    int g = i >> 7, mcol = i & 127;
    float cnt = fmaxf(gcnt[g], 1.0f);
    hg[g * 256 + mcol]       = gsum[i] / cnt;   // mean half
    hg[g * 256 + 128 + mcol] = gmax[i];         // max half
}

extern "C" void kernel_launch(void* const* d_in, const int* in_sizes, int n_in,
                              void* d_out, int out_size, void* d_ws, size_t ws_size,
                              hipStream_t stream) {
    (void)in_sizes; (void)n_in; (void)out_size; (void)ws_size;

    const float*     x     = (const float*)d_in[0];
    const long long* ei    = (const long long*)d_in[1];
    const long long* batch = (const long long*)d_in[2];
    const float*     W_in  = (const float*)d_in[3];
    const float*     b_in  = (const float*)d_in[4];
    const float*     W1    = (const float*)d_in[5];
    const float*     b1    = (const float*)d_in[6];
    const float*     W2    = (const float*)d_in[7];
    const float*     b2    = (const float*)d_in[8];
    const float*     Wp    = (const float*)d_in[9];
    const float*     bp    = (const float*)d_in[10];

    const long long* src = ei;            // edge_index[0]
    const long long* dst = ei + N_EDGES;  // edge_index[1]

    // Workspace carve-out (~77 MB), 256B aligned slices.
    char* ws = (char*)d_ws;
    size_t off = 0;
    auto alloc = [&](size_t bytes) -> float* {
        float* p = (float*)(ws + off);
        off += (bytes + 255) & ~(size_t)255;
        return p;
    };
    float* dinv = alloc((size_t)N_NODES * 4);            // deg -> rsqrt(deg)
    float* hb0  = alloc((size_t)N_NODES * 128 * 4);      // h0 / hw2
    float* hb1  = alloc((size_t)N_NODES * 128 * 4);      // hw1 / h2
    float* hb2  = alloc((size_t)N_NODES * 128 * 4);      // h1
    float* gsum = alloc((size_t)N_GRAPHS * 128 * 4);
    float* gmax = alloc((size_t)N_GRAPHS * 128 * 4);     // contiguous after gsum
    float* gcnt = alloc((size_t)N_GRAPHS * 4);           // contiguous after gmax
    float* hg   = alloc((size_t)N_GRAPHS * 256 * 4);

    const int B = 256;
    // --- degree / normalization ---
    fill_f32<<<(N_NODES + B - 1) / B, B, 0, stream>>>(dinv, 1.0f, N_NODES); // self-loop
    edge_deg<<<(N_EDGES + B - 1) / B, B, 0, stream>>>(dst, dinv, N_EDGES);
    to_rsqrt<<<(N_NODES + B - 1) / B, B, 0, stream>>>(dinv, N_NODES);

    // --- input projection ---
    input_proj<<<(N_NODES * 64 + B - 1) / B, B, 0, stream>>>(x, W_in, b_in, hb0, N_NODES);

    // --- GCN layer 1: hw = h0 @ W1 ; aggregate ; relu ---
    { dim3 g(N_NODES / 16, 128 / 64);
      wmma_gemm_f32<<<g, 128, 0, stream>>>(hb0, W1, nullptr, hb1, 64, 128, 0); }
    init_agg<<<(N_NODES * 128) / B, B, 0, stream>>>(hb2, hb1, b1, dinv, N_NODES * 128);
    edge_scatter<<<(N_EDGES * 32) / B, B, 0, stream>>>(src, dst, dinv, hb1, hb2, N_EDGES);
    relu_ip<<<(N_NODES * 128) / B, B, 0, stream>>>(hb2, N_NODES * 128);

    // --- GCN layer 2: hw = h1 @ W2 ; aggregate ; relu ---
    { dim3 g(N_NODES / 16, 128 / 64);
      wmma_gemm_f32<<<g, 128, 0, stream>>>(hb2, W2, nullptr, hb0, 128, 128, 0); }
    init_agg<<<(N_NODES * 128) / B, B, 0, stream>>>(hb1, hb0, b2, dinv, N_NODES * 128);
    edge_scatter<<<(N_EDGES * 32) / B, B, 0, stream>>>(src, dst, dinv, hb0, hb1, N_EDGES);
    relu_ip<<<(N_NODES * 128) / B, B, 0, stream>>>(hb1, N_NODES * 128);

    // --- pooling (gsum|gmax|gcnt are contiguous: single zero pass) ---
    int zcount = N_GRAPHS * 128 * 2 + N_GRAPHS;
    fill_f32<<<(zcount + B - 1) / B, B, 0, stream>>>(gsum, 0.0f, zcount);
    pool<<<(N_NODES * 128) / B, B, 0, stream>>>(hb1, batch, gsum, gmax, gcnt, N_NODES * 128);
    finalize_hg<<<(N_GRAPHS * 128 + B - 1) / B, B, 0, stream>>>(gsum, gmax, gcnt, hg);

    // --- final projection: out = relu(hg @ Wp + bp), fused epilogue ---
    { dim3 g(N_GRAPHS / 16, 256 / 64);
      wmma_gemm_f32<<<g, 128, 0, stream>>>(hg, Wp, bp, (float*)d_out, 256, 256, 1); }
}